// Router_73134703117019
// MI455X (gfx1250) — compile-verified
//
#include <hip/hip_runtime.h>

typedef float v2f __attribute__((ext_vector_type(2)));
typedef float v4f __attribute__((ext_vector_type(4)));
typedef float v8f __attribute__((ext_vector_type(8)));

#define HIDDEN 4096
#define NEXP 8
#define KSEG 4                 // K split across 4 waves per tile
#define TILES_PER_BLOCK 2      // 8 waves/block = 2 tiles x 4 K-segments
#define TOKENS_PER_BLOCK 32

__global__ __launch_bounds__(256) void moe_router_kernel(
    const float* __restrict__ hidden,
    const float* __restrict__ W,
    const float* __restrict__ bias,
    float* __restrict__ out,
    int n_tokens)
{
    __shared__ float s_part[TILES_PER_BLOCK][KSEG][16][NEXP];

    const int tid  = threadIdx.x;
    const int lane = tid & 31;
    const int wave = tid >> 5;
    const int tile = wave >> 2;   // 0..1 : which 16-token tile
    const int kseg = wave & 3;    // 0..3 : which K segment
    const int row  = lane & 15;   // token row (A) / expert col (B)
    const int half = lane >> 4;   // 0 or 1 (lane half)

    const int tileBase = blockIdx.x * TOKENS_PER_BLOCK + tile * 16;
    const int kbase = kseg * (HIDDEN / KSEG);

    // Clamp row for safety on a partial last block (T=16384 is exact anyway).
    int arow = tileBase + row;
    if (arow >= n_tokens) arow = n_tokens - 1;

    const float* pA = hidden + (size_t)arow * HIDDEN + kbase + 4 * half;
    const int col = row;
    const float bmask = (col < NEXP) ? 1.0f : 0.0f;            // zero-pad experts 8..15
    const float* pB = W + (size_t)(col & (NEXP - 1)) * HIDDEN + kbase + 4 * half;

    // Per 16-K step: lane loads float4 at k+4*half and k+8+4*half for A and B.
    // WMMA j pairs A/B K-values {k+c, k+c+1, k+c+4, k+c+5}, c = 8*(j/2) + 2*(j%2); union over the
    // 4 WMMAs covers k..k+15 exactly once, consistently between A and B
    // (K-order permutation is legal for a dot product).
    v8f acc = {};
    #pragma unroll 2
    for (int k = 0; k < HIDDEN / KSEG; k += 16) {
        v4f a0 = *(const v4f*)(pA + k);
        v4f a1 = *(const v4f*)(pA + k + 8);
        v4f b0 = *(const v4f*)(pB + k);
        v4f b1 = *(const v4f*)(pB + k + 8);
        b0 *= bmask;
        b1 *= bmask;
        v2f a, b;
        a.x = a0.x; a.y = a0.y; b.x = b0.x; b.y = b0.y;
        acc = __builtin_amdgcn_wmma_f32_16x16x4_f32(false, a, false, b, (short)0, acc, false, false);
        a.x = a0.z; a.y = a0.w; b.x = b0.z; b.y = b0.w;
        acc = __builtin_amdgcn_wmma_f32_16x16x4_f32(false, a, false, b, (short)0, acc, false, false);
        a.x = a1.x; a.y = a1.y; b.x = b1.x; b.y = b1.y;
        acc = __builtin_amdgcn_wmma_f32_16x16x4_f32(false, a, false, b, (short)0, acc, false, false);
        a.x = a1.z; a.y = a1.w; b.x = b1.z; b.y = b1.w;
        acc = __builtin_amdgcn_wmma_f32_16x16x4_f32(false, a, false, b, (short)0, acc, false, false);
    }

    // C layout: acc[r] holds (M = r + 8*half, N = lane&15). Dump partial 16x8.
    if (col < NEXP) {
        #pragma unroll
        for (int r = 0; r < 8; ++r) {
            int m = r + 8 * half;
            s_part[tile][kseg][m][col] = acc[r];
        }
    }
    __syncthreads();

    // Epilogue: one thread per token (32 per block).
    if (tid < TOKENS_PER_BLOCK) {
        const int tl = tid;
        const int token = blockIdx.x * TOKENS_PER_BLOCK + tl;
        const int ttile = tl >> 4;
        const int trow = tl & 15;
        if (token < n_tokens) {
            float logit[NEXP];
            #pragma unroll
            for (int e = 0; e < NEXP; ++e) {
                float s = bias[e];
                #pragma unroll
                for (int ks = 0; ks < KSEG; ++ks) s += s_part[ttile][ks][trow][e];
                logit[e] = s;
            }

            float* out_idx = out;
            float* out_w   = out + 2 * (size_t)n_tokens;
            float* out_lg  = out + 4 * (size_t)n_tokens;

            #pragma unroll
            for (int e = 0; e < NEXP; ++e)
                out_lg[(size_t)token * NEXP + e] = logit[e];

            // softmax over 8
            float mx = logit[0];
            #pragma unroll
            for (int e = 1; e < NEXP; ++e) mx = fmaxf(mx, logit[e]);
            float p[NEXP];
            float sum = 0.0f;
            #pragma unroll
            for (int e = 0; e < NEXP; ++e) { p[e] = expf(logit[e] - mx); sum += p[e]; }
            float inv = 1.0f / sum;
            #pragma unroll
            for (int e = 0; e < NEXP; ++e) p[e] *= inv;

            // top-2, first-occurrence wins ties (matches lax.top_k ordering)
            int i0 = 0; float p0 = p[0];
            #pragma unroll
            for (int e = 1; e < NEXP; ++e) if (p[e] > p0) { p0 = p[e]; i0 = e; }
            int i1 = -1; float p1 = -1.0f;
            #pragma unroll
            for (int e = 0; e < NEXP; ++e) if (e != i0 && p[e] > p1) { p1 = p[e]; i1 = e; }

            out_idx[(size_t)token * 2 + 0] = (float)i0;
            out_idx[(size_t)token * 2 + 1] = (float)i1;
            out_w[(size_t)token * 2 + 0]   = p0;
            out_w[(size_t)token * 2 + 1]   = p1;
        }
    }
}

extern "C" void kernel_launch(void* const* d_in, const int* in_sizes, int n_in,
                              void* d_out, int out_size, void* d_ws, size_t ws_size,
                              hipStream_t stream) {
    const float* hidden = (const float*)d_in[0];   // [4,4096,4096] f32
    const float* W      = (const float*)d_in[1];   // [8,4096] f32
    const float* b      = (const float*)d_in[2];   // [8] f32
    float* out = (float*)d_out;

    const int n_tokens = in_sizes[0] / HIDDEN;     // 16384
    const int blocks = (n_tokens + TOKENS_PER_BLOCK - 1) / TOKENS_PER_BLOCK;  // 512

    moe_router_kernel<<<blocks, 256, 0, stream>>>(hidden, W, b, out, n_tokens);
}